// LDDMMVariational_9474697855459
// MI455X (gfx1250) — compile-verified
//
#include <hip/hip_runtime.h>

// LDDMM variational shooting RHS, Gaussian kernel, sigma = 0.1.
// d_in[0] = mom (8192x3 f32), d_in[1] = control_points (8192x3 f32)
// d_out   = [dmom (8192x3) | dcp (8192x3)] f32
//
// MI455X strategy: whole problem is L2/LDS resident (192KB); cost is N^2=67M
// pair interactions (1 v_exp_f32 + ~11 VALU each). The two pairwise Gram tiles
// (x_i.x_j, p_i.p_j) go to the matrix pipe via V_WMMA_F32_16X16X4_F32 (D=3
// padded to K=4; f32 needed since the exponent is 50*d2 -- fp16 dot error
// would give ~10% kernel error). Split-K over j (8 slices) lifts wave count
// from 512 to 4096 so the WGP array is actually filled; partials are combined
// with global f32 atomics into a zeroed output. exp uses the raw v_exp_f32
// (args clamped to <=0; sub-denormal kernel values are negligible vs O(1)
// accumulators, so the libm ldexp fixup is dead weight).

typedef __attribute__((ext_vector_type(2))) float v2f;
typedef __attribute__((ext_vector_type(8))) float v8f;

#define SIG2INV 100.0f                  // 1/sigma^2
#define LOG2E   1.4426950408889634f
#define C_DOT   (100.0f * LOG2E)        // coefficient of x_i.x_j in exp2 arg
#define C_SQ    (-50.0f * LOG2E)        // coefficient of |x|^2 terms
#define SPLIT   8                       // j-slices (grid.y)
#define CHUNK   1024                    // N / SPLIT, staged in LDS (24KB)

#if __has_builtin(__builtin_amdgcn_exp2f)
#define FAST_EXP2(x) __builtin_amdgcn_exp2f(x)
#else
#define FAST_EXP2(x) __expf((x) * 0.6931471805599453f)
#endif

__global__ void zero_out_kernel(float* __restrict__ out, int n) {
    const int i = blockIdx.x * blockDim.x + threadIdx.x;
    if (i < n) out[i] = 0.0f;
}

__launch_bounds__(256)
__global__ void lddmm_wmma_kernel(const float* __restrict__ mom,
                                  const float* __restrict__ xg,
                                  float* __restrict__ out,
                                  int N)
{
    __shared__ float sx[CHUNK * 3];
    __shared__ float sp[CHUNK * 3];

    const int tid  = threadIdx.x;
    const int wave = tid >> 5;
    const int lane = tid & 31;
    const int half = lane >> 4;   // 0: K=0..1 / rows 0..7, 1: K=2..3 / rows 8..15
    const int l16  = lane & 15;

    const int i0    = (blockIdx.x * 8 + wave) * 16;  // this wave's 16 rows
    const int jbase = blockIdx.y * CHUNK;            // this block's j-slice

    // ---- stage this block's j-slice into LDS (x and clamped p)
    for (int t = tid; t < CHUNK * 3; t += 256) {
        const int g = jbase * 3 + t;
        sx[t] = xg[g];
        sp[t] = fminf(fmaxf(mom[g], -1.f), 1.f);
    }

    // ---- i-side WMMA A operands (16x4 f32 A layout: 2 VGPRs, K split by half)
    const int ii = i0 + l16;
    const float xi0 = xg[ii * 3 + 0];
    const float xi1 = xg[ii * 3 + 1];
    const float xi2 = xg[ii * 3 + 2];
    const float pi0 = fminf(fmaxf(mom[ii * 3 + 0], -1.f), 1.f);
    const float pi1 = fminf(fmaxf(mom[ii * 3 + 1], -1.f), 1.f);
    const float pi2 = fminf(fmaxf(mom[ii * 3 + 2], -1.f), 1.f);

    v2f aX, aP;
    aX.x = half ? xi2 : xi0;  aX.y = half ? 0.f : xi1;
    aP.x = half ? pi2 : pi0;  aP.y = half ? 0.f : pi1;

    // pre-scaled -50*log2e*|x_i|^2 for the 8 rows this lane's C-half covers
    float sqrowS[8];
#pragma unroll
    for (int r = 0; r < 8; ++r) {
        const int row = i0 + half * 8 + r;
        const float a = xg[row * 3 + 0];
        const float b = xg[row * 3 + 1];
        const float c = xg[row * 3 + 2];
        sqrowS[r] = C_SQ * (a * a + b * b + c * c);
    }

    float accRow[8], accWx0[8], accWx1[8], accWx2[8];
    float accKp0[8], accKp1[8], accKp2[8];
#pragma unroll
    for (int r = 0; r < 8; ++r) {
        accRow[r] = 0.f;
        accWx0[r] = accWx1[r] = accWx2[r] = 0.f;
        accKp0[r] = accKp1[r] = accKp2[r] = 0.f;
    }

    __syncthreads();

    for (int jt = 0; jt < CHUNK; jt += 16) {
        const int jj = jt + l16;              // this lane's column j
        const float xj0 = sx[jj * 3 + 0];
        const float xj1 = sx[jj * 3 + 1];
        const float xj2 = sx[jj * 3 + 2];
        const float pj0 = sp[jj * 3 + 0];
        const float pj1 = sp[jj * 3 + 1];
        const float pj2 = sp[jj * 3 + 2];
        const float sqjS = C_SQ * (xj0 * xj0 + xj1 * xj1 + xj2 * xj2);

        // B operands (4x16 f32 B layout mirrors A: K split by lane half)
        v2f bX, bP;
        bX.x = half ? xj2 : xj0;  bX.y = half ? 0.f : xj1;
        bP.x = half ? pj2 : pj0;  bP.y = half ? 0.f : pj1;

        v8f z = {0.f, 0.f, 0.f, 0.f, 0.f, 0.f, 0.f, 0.f};
        v8f xdot = __builtin_amdgcn_wmma_f32_16x16x4_f32(
            false, aX, false, bX, (short)0, z, false, false);
        v8f pdot = __builtin_amdgcn_wmma_f32_16x16x4_f32(
            false, aP, false, bP, (short)0, z, false, false);

#pragma unroll
        for (int r = 0; r < 8; ++r) {
            // exp2-domain arg: min(.,0) == the reference's max(d2,0) clamp
            const float arg = fminf(fmaf(xdot[r], C_DOT, sqrowS[r] + sqjS), 0.0f);
            const float K   = FAST_EXP2(arg);      // bare v_exp_f32
            const float W   = K * pdot[r];
            accRow[r] += W;
            accWx0[r] += W * xj0;  accWx1[r] += W * xj1;  accWx2[r] += W * xj2;
            accKp0[r] += K * pj0;  accKp1[r] += K * pj1;  accKp2[r] += K * pj2;
        }
    }

    // reduce over the 16 lanes of each half (xor masks 1,2,4,8 stay in-half)
#pragma unroll
    for (int m = 1; m <= 8; m <<= 1) {
#pragma unroll
        for (int r = 0; r < 8; ++r) {
            accRow[r] += __shfl_xor(accRow[r], m);
            accWx0[r] += __shfl_xor(accWx0[r], m);
            accWx1[r] += __shfl_xor(accWx1[r], m);
            accWx2[r] += __shfl_xor(accWx2[r], m);
            accKp0[r] += __shfl_xor(accKp0[r], m);
            accKp1[r] += __shfl_xor(accKp1[r], m);
            accKp2[r] += __shfl_xor(accKp2[r], m);
        }
    }

    if (l16 == 0) {
        const int base = i0 + half * 8;
#pragma unroll
        for (int r = 0; r < 8; ++r) {
            const int i = base + r;
            const float xr0 = xg[i * 3 + 0];
            const float xr1 = xg[i * 3 + 1];
            const float xr2 = xg[i * 3 + 2];
            // dmom partial = (1/sig^2) * (x_i * rowsum(W) - W@x)
            atomicAdd(&out[i * 3 + 0], SIG2INV * (xr0 * accRow[r] - accWx0[r]));
            atomicAdd(&out[i * 3 + 1], SIG2INV * (xr1 * accRow[r] - accWx1[r]));
            atomicAdd(&out[i * 3 + 2], SIG2INV * (xr2 * accRow[r] - accWx2[r]));
            // dcp partial = K @ p
            atomicAdd(&out[N * 3 + i * 3 + 0], accKp0[r]);
            atomicAdd(&out[N * 3 + i * 3 + 1], accKp1[r]);
            atomicAdd(&out[N * 3 + i * 3 + 2], accKp2[r]);
        }
    }
}

extern "C" void kernel_launch(void* const* d_in, const int* in_sizes, int n_in,
                              void* d_out, int out_size, void* d_ws, size_t ws_size,
                              hipStream_t stream) {
    (void)n_in; (void)d_ws; (void)ws_size;
    const float* mom = (const float*)d_in[0];
    const float* cp  = (const float*)d_in[1];
    float* out = (float*)d_out;
    const int N = in_sizes[0] / 3;          // 8192

    zero_out_kernel<<<(out_size + 255) / 256, 256, 0, stream>>>(out, out_size);

    dim3 grid(N / (8 * 16), SPLIT);         // 64 i-blocks x 8 j-slices
    lddmm_wmma_kernel<<<grid, 256, 0, stream>>>(mom, cp, out, N);
}